// RnRoPE_42958262894757
// MI455X (gfx1250) — compile-verified
//
#include <hip/hip_runtime.h>

// RnRoPE fused: out = RoPE(x @ Q^T) @ Q for q and k, Q = Householder product.
// f32 WMMA (V_WMMA_F32_16X16X4_F32) keeps exact reference precision.
// v2: dual independent WMMA accumulator chains (RoPE pair t / t+4), 256-thread
// blocks (8 wave32), lower VGPR pressure -> better WGP occupancy.

typedef __attribute__((ext_vector_type(2))) float v2f;
typedef __attribute__((ext_vector_type(4))) float v4f;
typedef __attribute__((ext_vector_type(8))) float v8f;

#define DIM   128
#define NREFL 64
#define SEQ   8192
#define NROWS (2 * 16 * 8192)            // B*H*S rows per tensor = 262144
#define QS    132                         // padded LDS row stride (dwords)
#define WAVES 8
#define ROWS_PER_BLOCK (WAVES * 16)       // 128
#define NBLOCKS ((2 * NROWS) / ROWS_PER_BLOCK)  // 4096
#define GRID  2048

// ---------------------------------------------------------------------------
// Kernel 1: build Q (128x128) from 64 Householder reflections.
// One block, 128 threads; thread t owns column t of Q -> rank-1 update is
// column-private, so the whole scan needs no barriers.
// ---------------------------------------------------------------------------
__global__ __launch_bounds__(128) void build_householder_q(
    const float* __restrict__ vs, float* __restrict__ qout)
{
    __shared__ float Q[DIM * DIM];       // 64 KB
    const int t = threadIdx.x;           // column owner

    for (int d = 0; d < DIM; ++d) Q[d * DIM + t] = (d == t) ? 1.0f : 0.0f;

    for (int r = 0; r < NREFL; ++r) {
        const float* v = vs + r * DIM;   // uniform across lanes -> scalar loads
        float w = 0.0f, vn = 0.0f;
        for (int d = 0; d < DIM; ++d) {
            float vd = v[d];
            w  += vd * Q[d * DIM + t];   // (v @ Q)[t] : column-t reads only
            vn += vd * vd;
        }
        float sw = (2.0f / (vn + 1e-8f)) * w;
        for (int d = 0; d < DIM; ++d) Q[d * DIM + t] -= v[d] * sw;
    }
    for (int d = 0; d < DIM; ++d) qout[d * DIM + t] = Q[d * DIM + t];
}

// ---------------------------------------------------------------------------
// Kernel 2: fused  (x @ Q^T) -> RoPE -> (@ Q)  via f32 WMMA 16x16x4.
// 256 threads = 8 wave32; each wave independently processes 16 rows/iter.
// ---------------------------------------------------------------------------
__global__ __launch_bounds__(256) void rnrope_fused(
    const float* __restrict__ qin,  const float* __restrict__ kin,
    const float* __restrict__ cosp, const float* __restrict__ sinp,
    const float* __restrict__ qmat,
    float* __restrict__ outq, float* __restrict__ outk)
{
    __shared__ float sQ[DIM * QS];             // 67584 B, Q row-major padded
    __shared__ float sStage[WAVES][16 * QS];   // 67584 B, wave-private tiles

    const int tid  = threadIdx.x;
    const int wv   = tid >> 5;
    const int lane = tid & 31;
    const int m    = lane & 15;           // tile row (A) / tile col (B,C)
    const int hi   = lane >> 4;           // lane half
    const int j0   = hi << 1;             // k-subindex base: 0 or 2

    // One-time cooperative load of Q into LDS (v4f, coalesced).
    for (int i = tid; i < DIM * (DIM / 4); i += 256) {
        int row = i >> 5;
        int c4  = (i & 31) << 2;
        v4f val = *(const v4f*)(qmat + row * DIM + c4);
        *(v4f*)(&sQ[row * QS + c4]) = val;
    }
    __syncthreads();

    float* st = &sStage[wv][0];

    for (int bi = blockIdx.x; bi < NBLOCKS; bi += gridDim.x) {
        const int grow = bi * ROWS_PER_BLOCK;              // [0, 2*NROWS)
        const int isK  = grow >= NROWS;
        const float* src = isK ? kin  : qin;
        float*       dst = isK ? outk : outq;
        const int lrow  = (grow - (isK ? NROWS : 0)) + wv * 16; // tensor-local
        const int sbase = lrow & (SEQ - 1);                // position index s

        // ---- stage 16x128 input rows into wave-private LDS (coalesced NT) ----
        const float* xin = src + (size_t)lrow * DIM;
        #pragma unroll
        for (int r = 0; r < 16; ++r) {
            v4f val = __builtin_nontemporal_load(
                (const v4f*)(xin + r * DIM + (lane << 2)));
            *(v4f*)(&st[r * QS + (lane << 2)]) = val;
        }

        // ---- A fragments for GEMM1: A[m][k] = X[m][4kk + k] ----
        v2f a[32];
        #pragma unroll
        for (int kk = 0; kk < 32; ++kk)
            a[kk] = *(const v2f*)(&st[m * QS + (kk << 2) + j0]);

        // ---- GEMM1 + RoPE, one N-tile pair (t, t+4) at a time.
        //      Two independent WMMA chains per pair -> no serial C dependency.
        #pragma unroll
        for (int t = 0; t < 4; ++t) {
            v8f c0 = {};   // N-tile t      (cols t*16   .. t*16+15)
            v8f c1 = {};   // N-tile t + 4  (cols t*16+64.. )
            #pragma unroll
            for (int kk = 0; kk < 32; ++kk) {
                // B[k][n] = Q[nt*16+n][4kk+k]; lane n=m, k in {j0, j0+1}
                v2f b0 = *(const v2f*)(&sQ[(t * 16 + m) * QS + (kk << 2) + j0]);
                v2f b1 = *(const v2f*)(&sQ[((t + 4) * 16 + m) * QS + (kk << 2) + j0]);
                c0 = __builtin_amdgcn_wmma_f32_16x16x4_f32(
                        false, a[kk], false, b0, (short)0, c0, false, false);
                c1 = __builtin_amdgcn_wmma_f32_16x16x4_f32(
                        false, a[kk], false, b1, (short)0, c1, false, false);
            }
            // RoPE: cols n and n+64 are exactly (c0, c1) at same lane/vgpr;
            // cos/sin periodic in d with period 64 -> one load pair serves both.
            #pragma unroll
            for (int r = 0; r < 8; ++r) {
                int mm   = r + (hi << 3);
                int srow = sbase + mm;
                float cv = cosp[srow * DIM + t * 16 + m];
                float sv = sinp[srow * DIM + t * 16 + m];
                float x0 = c0[r], x1 = c1[r];
                st[mm * QS + t * 16 + m]       = x0 * cv - x1 * sv; // n < 64
                st[mm * QS + (t + 4) * 16 + m] = x1 * cv + x0 * sv; // n >= 64
            }
        }

        // ---- A fragments for GEMM2 from rotated stage ----
        #pragma unroll
        for (int kk = 0; kk < 32; ++kk)
            a[kk] = *(const v2f*)(&st[m * QS + (kk << 2) + j0]);

        // ---- GEMM2: out[m][e] = sum_d rot[m][d] * Q[d][e], paired chains ----
        #pragma unroll
        for (int t = 0; t < 4; ++t) {
            v8f c0 = {};
            v8f c1 = {};
            #pragma unroll
            for (int kk = 0; kk < 32; ++kk) {
                // B[k][n] = Q[4kk+k][nt*16+n]
                v2f b0, b1;
                b0.x = sQ[((kk << 2) + j0    ) * QS + t * 16 + m];
                b0.y = sQ[((kk << 2) + j0 + 1) * QS + t * 16 + m];
                b1.x = sQ[((kk << 2) + j0    ) * QS + (t + 4) * 16 + m];
                b1.y = sQ[((kk << 2) + j0 + 1) * QS + (t + 4) * 16 + m];
                c0 = __builtin_amdgcn_wmma_f32_16x16x4_f32(
                        false, a[kk], false, b0, (short)0, c0, false, false);
                c1 = __builtin_amdgcn_wmma_f32_16x16x4_f32(
                        false, a[kk], false, b1, (short)0, c1, false, false);
            }
            #pragma unroll
            for (int r = 0; r < 8; ++r) {
                int mm = r + (hi << 3);
                float* drow = dst + (size_t)(lrow + mm) * DIM;
                __builtin_nontemporal_store(c0[r], drow + t * 16 + m);
                __builtin_nontemporal_store(c1[r], drow + (t + 4) * 16 + m);
            }
        }
    }
}

// ---------------------------------------------------------------------------
extern "C" void kernel_launch(void* const* d_in, const int* in_sizes, int n_in,
                              void* d_out, int out_size, void* d_ws, size_t ws_size,
                              hipStream_t stream)
{
    (void)in_sizes; (void)n_in; (void)out_size; (void)ws_size;
    const float* q    = (const float*)d_in[0];
    const float* k    = (const float*)d_in[1];
    const float* vs   = (const float*)d_in[2];
    const float* cosp = (const float*)d_in[3];
    const float* sinp = (const float*)d_in[4];

    float* qws  = (float*)d_ws;                  // 128*128 f32 = 64 KB scratch
    float* outq = (float*)d_out;                 // q_final
    float* outk = outq + (size_t)NROWS * DIM;    // k_final (concatenated)

    build_householder_q<<<1, 128, 0, stream>>>(vs, qws);
    rnrope_fused<<<GRID, 256, 0, stream>>>(q, k, cosp, sinp, qws, outq, outk);
}